// one_step_attention_90177133347145
// MI455X (gfx1250) — compile-verified
//
#include <hip/hip_runtime.h>
#include <math.h>

typedef __attribute__((ext_vector_type(2))) float v2f;
typedef __attribute__((ext_vector_type(8))) float v8f;

#define B_   16
#define T_   4096
#define DA_  1024
#define DS_  1024
#define H_   10
#define HP_  16      // H padded to WMMA N=16
#define TS_  128     // t-rows per block (8 waves x 16)
#define KC_  32      // K chunk staged in LDS
#define LSTR 36      // LDS row stride (floats): bank-conflict pad, 16B-aligned
#define TCH_ 4       // t-chunks for ctx pass

#ifndef __has_builtin
#define __has_builtin(x) 0
#endif
#if __has_builtin(__builtin_amdgcn_global_load_async_to_lds_b128) && \
    __has_builtin(__builtin_amdgcn_s_wait_asynccnt)
#define USE_ASYNC_LDS 1
#else
#define USE_ASYNC_LDS 0
#warning "gfx1250 async-to-LDS builtins unavailable; using synchronous LDS staging fallback"
#endif

// 16-byte int vector in the address spaces the async builtin expects:
// param0: int4 AS1 (printed as 'int4 __device__*'), param1: int4 AS3.
typedef __attribute__((__vector_size__(16))) int v4i_t;
typedef __attribute__((address_space(1))) v4i_t as1_v4i;
typedef __attribute__((address_space(3))) v4i_t as3_v4i;

// ---------- kernel 1: pad W1[0:1024, 0:10] -> wpad[1024][16] (zero cols 10..15)
__global__ __launch_bounds__(256)
void pad_w_kernel(const float* __restrict__ W1, float* __restrict__ wpad) {
  int idx = blockIdx.x * 256 + threadIdx.x;   // 0..16383
  int d = idx >> 4, h = idx & 15;
  wpad[idx] = (h < H_) ? W1[d * H_ + h] : 0.f;
}

// ---------- kernel 2: c[b][h] = b1[h] + sum_d s_prev[b,d] * W1[DA+d, h]
__global__ __launch_bounds__(256)
void sconst_kernel(const float* __restrict__ s_prev, const float* __restrict__ W1,
                   const float* __restrict__ b1, float* __restrict__ cws) {
  __shared__ float sh[256];
  int b = blockIdx.x;
  int h = threadIdx.x & 15;
  int chunk = threadIdx.x >> 4;               // 16 chunks of 64 d's
  float p = 0.f;
  if (h < H_) {
    #pragma unroll 4
    for (int i = 0; i < 64; ++i) {
      int d = chunk * 64 + i;
      p += s_prev[b * DS_ + d] * W1[(DA_ + d) * H_ + h];
    }
  }
  sh[threadIdx.x] = p;
  __syncthreads();
  if (threadIdx.x < 16) {
    float s = 0.f;
    #pragma unroll
    for (int c = 0; c < 16; ++c) s += sh[c * 16 + threadIdx.x];
    cws[b * HP_ + threadIdx.x] = s + ((threadIdx.x < H_) ? b1[threadIdx.x] : 0.f);
  }
}

// Stage a[b, t0..t0+127, kc..kc+31] into an LDS buffer (coalesced 16B per op).
__device__ __forceinline__
void fill_chunk(const float* __restrict__ a, float* buf,
                int b, int t0, int kc, int tid) {
  #pragma unroll
  for (int i = 0; i < 4; ++i) {
    int f   = i * 256 + tid;          // float4 id: 128 rows * 8 float4
    int row = f >> 3;
    int c4  = (f & 7) << 2;
    const float* gp = a + ((size_t)(b * T_ + t0 + row)) * DA_ + kc + c4;
    float*       lp = &buf[row * LSTR + c4];
#if USE_ASYNC_LDS
    __builtin_amdgcn_global_load_async_to_lds_b128(
        (as1_v4i*)(size_t)gp, (as3_v4i*)(unsigned)(size_t)lp, 0, 0);
#else
    *reinterpret_cast<float4*>(lp) = *reinterpret_cast<const float4*>(gp);
#endif
  }
}

// ---------- kernel 3: e[b,t] = relu( tanh(a[b,t,:]@W1a + c[b]) @ W2 + b2 )
// WMMA f32 16x16x4: wave handles 16 t-rows, K stepped by 4 over DA=1024.
// Double-buffered LDS; async DMA of chunk k+1 overlaps WMMA on chunk k.
__global__ __launch_bounds__(256)
void score_kernel(const float* __restrict__ a, const float* __restrict__ wpad,
                  const float* __restrict__ cws, const float* __restrict__ W2,
                  const float* __restrict__ b2, float* __restrict__ ews) {
  __shared__ float As[2 * TS_ * LSTR];
  const int tid  = threadIdx.x;
  const int wave = tid >> 5;
  const int lane = tid & 31;
  const int b    = blockIdx.y;
  const int t0   = blockIdx.x * TS_;
  const int n    = lane & 15;      // N index (h) / row-within-16
  const int half = lane >> 4;      // 0: K=0,1   1: K=2,3  (A 16x4 layout)

  v8f acc = {};
  const int rowLds = wave * 16 + n;

  fill_chunk(a, As, b, t0, 0, tid);     // prime buffer 0

  for (int kci = 0; kci < DA_ / KC_; ++kci) {
    const int kc = kci * KC_;
    float* cur = As + (kci & 1) * (TS_ * LSTR);
    float* nxt = As + ((kci & 1) ^ 1) * (TS_ * LSTR);

#if USE_ASYNC_LDS
    __builtin_amdgcn_s_wait_asynccnt(0);   // current buffer's DMA done (per wave)
#endif
    __syncthreads();                        // visible to all waves; prior compute done

    if (kc + KC_ < DA_)
      fill_chunk(a, nxt, b, t0, kc + KC_, tid);  // overlaps with WMMA below

    #pragma unroll
    for (int k = 0; k < KC_; k += 4) {
      // A fragment: lane<16 -> (K=k,k+1), lane>=16 -> (K=k+2,k+3), row = rowLds
      const int kk = k + half * 2;
      const float2 av = *reinterpret_cast<const float2*>(&cur[rowLds * LSTR + kk]);
      v2f afrag; afrag[0] = av.x; afrag[1] = av.y;
      // B fragment: 4x16 rows striped across lanes per VGPR
      const int kg = kc + k + half * 2;
      v2f bfrag;
      bfrag[0] = wpad[kg * HP_ + n];
      bfrag[1] = wpad[(kg + 1) * HP_ + n];
      acc = __builtin_amdgcn_wmma_f32_16x16x4_f32(
          false, afrag, false, bfrag, (short)0, acc, false, false);
    }
  }

  // epilogue: D layout -> VGPR j: lanes0-15 M=j, lanes16-31 M=j+8; N = lane&15
  const float cv    = cws[b * HP_ + n];
  const float w2v   = (n < H_) ? W2[n] : 0.f;
  const float bias2 = b2[0];
  #pragma unroll
  for (int j = 0; j < 8; ++j) {
    float hv = tanhf(acc[j] + cv) * w2v;
    hv += __shfl_xor(hv, 1, 32);
    hv += __shfl_xor(hv, 2, 32);
    hv += __shfl_xor(hv, 4, 32);
    hv += __shfl_xor(hv, 8, 32);    // sum over N within each 16-lane half
    float ev = fmaxf(hv + bias2, 0.f);
    if (n == 0) {
      int t = t0 + wave * 16 + j + half * 8;
      ews[b * T_ + t] = ev;
    }
  }
}

// ---------- kernel 4: per-b softmax stats (max, 1/sum exp)
__global__ __launch_bounds__(256)
void softmax_stats_kernel(const float* __restrict__ ews, float* __restrict__ stats) {
  __shared__ float red[256];
  int b = blockIdx.x, tid = threadIdx.x;
  float mx = -3.4e38f;
  for (int i = tid; i < T_; i += 256) mx = fmaxf(mx, ews[b * T_ + i]);
  red[tid] = mx;
  __syncthreads();
  for (int s = 128; s > 0; s >>= 1) {
    if (tid < s) red[tid] = fmaxf(red[tid], red[tid + s]);
    __syncthreads();
  }
  mx = red[0];
  __syncthreads();
  float sum = 0.f;
  for (int i = tid; i < T_; i += 256) sum += __expf(ews[b * T_ + i] - mx);
  red[tid] = sum;
  __syncthreads();
  for (int s = 128; s > 0; s >>= 1) {
    if (tid < s) red[tid] += red[tid + s];
    __syncthreads();
  }
  if (tid == 0) { stats[b * 2] = mx; stats[b * 2 + 1] = 1.f / red[0]; }
}

// ---------- kernel 5a: partial ctx over a t-chunk: part[tc][b*DA+d]
__global__ __launch_bounds__(256)
void ctx_partial_kernel(const float* __restrict__ a, const float* __restrict__ ews,
                        const float* __restrict__ stats, float* __restrict__ part) {
  __shared__ float al[256];
  const int b = blockIdx.z, tc = blockIdx.y, tid = threadIdx.x;
  const int d = blockIdx.x * 256 + tid;
  const float mx  = stats[b * 2];
  const float inv = stats[b * 2 + 1];
  float accv = 0.f;
  const int tbeg = tc * (T_ / TCH_);
  const int tend = tbeg + (T_ / TCH_);
  for (int t0 = tbeg; t0 < tend; t0 += 256) {
    al[tid] = __expf(ews[b * T_ + t0 + tid] - mx) * inv;
    __syncthreads();
    if (t0 + 256 < tend)
      __builtin_prefetch(a + ((size_t)(b * T_ + t0 + 256)) * DA_ + d, 0, 0);
    #pragma unroll 8
    for (int tt = 0; tt < 256; ++tt)
      accv += al[tt] * a[((size_t)(b * T_ + t0 + tt)) * DA_ + d];
    __syncthreads();
  }
  part[((size_t)tc * B_ * DA_) + b * DA_ + d] = accv;
}

// ---------- kernel 5b: reduce partials -> out[b*DA+d]
__global__ __launch_bounds__(256)
void ctx_reduce_kernel(const float* __restrict__ part, float* __restrict__ out) {
  int i = blockIdx.x * 256 + threadIdx.x;   // 0..16383
  float s = 0.f;
  #pragma unroll
  for (int tc = 0; tc < TCH_; ++tc) s += part[(size_t)tc * B_ * DA_ + i];
  out[i] = s;
}

extern "C" void kernel_launch(void* const* d_in, const int* in_sizes, int n_in,
                              void* d_out, int out_size, void* d_ws, size_t ws_size,
                              hipStream_t stream) {
  (void)in_sizes; (void)n_in; (void)out_size; (void)ws_size;
  const float* a      = (const float*)d_in[0];
  const float* s_prev = (const float*)d_in[1];
  const float* W1     = (const float*)d_in[2];
  const float* b1     = (const float*)d_in[3];
  const float* W2     = (const float*)d_in[4];
  const float* b2     = (const float*)d_in[5];
  float* out = (float*)d_out;

  float* ws    = (float*)d_ws;
  float* wpad  = ws;                            // 1024*16 floats
  float* cws   = wpad + DA_ * HP_;              // 16*16 floats
  float* ews   = cws + B_ * HP_;                // 16*4096 floats
  float* stats = ews + B_ * T_;                 // 32 floats
  float* part  = stats + 2 * B_;                // TCH*16*1024 floats

  pad_w_kernel<<<(DA_ * HP_) / 256, 256, 0, stream>>>(W1, wpad);
  sconst_kernel<<<B_, 256, 0, stream>>>(s_prev, W1, b1, cws);
  score_kernel<<<dim3(T_ / TS_, B_), 256, 0, stream>>>(a, wpad, cws, W2, b2, ews);
  softmax_stats_kernel<<<B_, 256, 0, stream>>>(ews, stats);
  ctx_partial_kernel<<<dim3(DA_ / 256, TCH_, B_), 256, 0, stream>>>(a, ews, stats, part);
  ctx_reduce_kernel<<<(B_ * DA_) / 256, 256, 0, stream>>>(part, out);
}